// EGCN_71038759076269
// MI455X (gfx1250) — compile-verified
//
#include <hip/hip_runtime.h>
#include <hip/hip_bf16.h>
#include <math.h>

#define DIM 64

typedef float v2f __attribute__((ext_vector_type(2)));
typedef float v4f __attribute__((ext_vector_type(4)));
typedef float v8f __attribute__((ext_vector_type(8)));

// Hardware f32 atomic add, no-return form (STOREcnt path), device scope so the
// RMW is performed at the device-coherent L2. Inline asm guarantees the
// global_atomic_add_f32 instruction (no CAS-loop expansion possible).
__device__ __forceinline__ void atom_add_f32(float* p, float v) {
    asm volatile("global_atomic_add_f32 %0, %1, off scope:SCOPE_DEV"
                 :: "v"(p), "v"(v)
                 : "memory");
}

// ---------------------------------------------------------------------------
// Edge scatter. One 16-lane group per edge; each lane moves a float4 (b128
// gather of the 256B source row), scales by the edge weight, and scatters
// with 4 hardware f32 atomics into the destination row.
// ---------------------------------------------------------------------------
__global__ void __launch_bounds__(256)
k_scatter_edges(const float* __restrict__ xin, const long long* __restrict__ adj,
                const float* __restrict__ w, float* __restrict__ out, int E) {
    const int e   = (int)(blockIdx.x * (blockDim.x >> 4) + (threadIdx.x >> 4));
    const int sub = (int)(threadIdx.x & 15);
    if (e >= E) return;
    const int s = (int)adj[e];              // adj[0][e]
    const int d = (int)adj[(long)E + e];    // adj[1][e]
    const float we = w[e];
    v4f v = *((const v4f*)(xin + (long)s * DIM) + sub);
    v *= we;
    float* op = out + (long)d * DIM + sub * 4;
    atom_add_f32(op + 0, v.x);
    atom_add_f32(op + 1, v.y);
    atom_add_f32(op + 2, v.z);
    atom_add_f32(op + 3, v.w);
}

// ---------------------------------------------------------------------------
// Row L2-normalize via WMMA Gram-diagonal.
// One wave handles 16 rows (16x64 tile). norm^2 of the rows = diag(X * X^T),
// computed with 16 chained V_WMMA_F32_16X16X4_F32 (K=64 in chunks of 4).
// The 16x4 A layout and 4x16 B layout are lane/VGPR symmetric, so the same
// registers serve as both A and B (B == A^T content).
// ---------------------------------------------------------------------------
__global__ void __launch_bounds__(256)
k_normalize_wmma(const float* __restrict__ x, float* __restrict__ y, int nwaves) {
    const int wid  = (int)((blockIdx.x * blockDim.x + threadIdx.x) >> 5);
    const int lane = (int)(threadIdx.x & 31);
    if (wid >= nwaves) return;               // wave-uniform exit: EXEC stays all-1s
    const long base = (long)wid * 16;        // first row of this wave's tile
    const int  half = lane >> 4;             // 0 -> K%4 in {0,1}; 1 -> K%4 in {2,3}
    const int  m    = lane & 15;             // row within tile
    const float* rowp = x + (base + m) * DIM + 2 * half;

    v8f acc = {0.f, 0.f, 0.f, 0.f, 0.f, 0.f, 0.f, 0.f};
#pragma unroll
    for (int c = 0; c < 16; ++c) {
        v2f a = *(const v2f*)(rowp + c * 4);  // A tile chunk: K = 4c + {0,1}+2*half
        acc = __builtin_amdgcn_wmma_f32_16x16x4_f32(false, a, false, a,
                                                    (short)0, acc, false, false);
    }

    // Diagonal of the 16x16 f32 D tile:
    //   rows 0..7  -> lane r,    VGPR r
    //   rows 8..15 -> lane 16+r, VGPR r-8
    const int k = lane & 7;
    float d = acc[0];
    d = (k == 1) ? acc[1] : d;
    d = (k == 2) ? acc[2] : d;
    d = (k == 3) ? acc[3] : d;
    d = (k == 4) ? acc[4] : d;
    d = (k == 5) ? acc[5] : d;
    d = (k == 6) ? acc[6] : d;
    d = (k == 7) ? acc[7] : d;
    const float inv = 1.0f / fmaxf(sqrtf(d), 1e-12f);

    // Scale+store: per step, lanes 0-15 cover row 2i, lanes 16-31 cover row
    // 2i+1 (32 lanes x float4 = 128 floats = 2 rows). Fully coalesced b128.
    // Per-row scales are wave-uniform: extract with v_readlane (compile-time
    // source lanes) into SGPRs -- no LDS/ds_bpermute traffic.
#pragma unroll
    for (int i = 0; i < 8; ++i) {
        const int r0 = 2 * i, r1 = 2 * i + 1;
        const int h0 = (r0 < 8) ? r0 : 16 + r0;   // lane holding inv for row r0
        const int h1 = (r1 < 8) ? r1 : 16 + r1;   // lane holding inv for row r1
        const float s0 =
            __int_as_float(__builtin_amdgcn_readlane(__float_as_int(inv), h0));
        const float s1 =
            __int_as_float(__builtin_amdgcn_readlane(__float_as_int(inv), h1));
        const float s  = half ? s1 : s0;
        const v4f* sp = (const v4f*)(x + (base + r0) * DIM) + lane;
        v4f*       dp = (v4f*)(y + (base + r0) * DIM) + lane;
        v4f v = *sp;
        v *= s;
        *dp = v;
    }
}

// Scalar safety net for a row-count not divisible by 16 (unused at N=100000).
__global__ void k_normalize_tail(const float* __restrict__ x, float* __restrict__ y,
                                 int start, int N) {
    int r = start + (int)(blockIdx.x * blockDim.x + threadIdx.x);
    if (r >= N) return;
    const float* xr = x + (long)r * DIM;
    float s = 0.f;
#pragma clang loop unroll(disable)
    for (int c = 0; c < DIM; ++c) s += xr[c] * xr[c];
    const float inv = 1.0f / fmaxf(sqrtf(s), 1e-12f);
    float* yr = y + (long)r * DIM;
#pragma clang loop unroll(disable)
    for (int c = 0; c < DIM; ++c) yr[c] = xr[c] * inv;
}

// Utility kernels: float4 zero-fill and float4 accumulate.
__global__ void k_zero4(v4f* __restrict__ p, long n4) {
    long i = (long)blockIdx.x * blockDim.x + threadIdx.x;
    if (i < n4) p[i] = (v4f){0.f, 0.f, 0.f, 0.f};
}

__global__ void k_add4(v4f* __restrict__ out, const v4f* __restrict__ a, long n4) {
    long i = (long)blockIdx.x * blockDim.x + threadIdx.x;
    if (i < n4) out[i] += a[i];
}

// ---------------------------------------------------------------------------
// Orchestration (minimal workspace: d_ws holds only x1, N*DIM floats):
//   1) x0 = l2norm(emb)              -> d_out
//   2) ws = 0
//   3) ws  = scatter(x0)             (x1)
//   4) d_out += ws                   (x0 + x1)
//   5) d_out += scatter(ws)          (adds x2 contributions directly)
// ---------------------------------------------------------------------------
extern "C" void kernel_launch(void* const* d_in, const int* in_sizes, int n_in,
                              void* d_out, int out_size, void* d_ws, size_t ws_size,
                              hipStream_t stream) {
    const float*     emb = (const float*)d_in[0];
    const float*     w   = (const float*)d_in[1];
    const long long* adj = (const long long*)d_in[2];
    float* out = (float*)d_out;
    float* x1  = (float*)d_ws;

    const int  N  = in_sizes[0] / DIM;
    const int  E  = in_sizes[1];
    const long n4 = (long)N * DIM / 4;

    // 1) normalize -> d_out
    const int nwaves = N / 16;
    if (nwaves > 0) {
        const int thr = nwaves * 32;
        const int blk = (thr + 255) / 256;
        k_normalize_wmma<<<blk, 256, 0, stream>>>(emb, out, nwaves);
    }
    const int tail = N - nwaves * 16;
    if (tail > 0)
        k_normalize_tail<<<(tail + 63) / 64, 64, 0, stream>>>(emb, out, nwaves * 16, N);

    const int eblk = (E + 15) / 16;   // 16 edges per 256-thread block
    const int vblk = (int)((n4 + 255) / 256);

    // 2) ws = 0
    k_zero4<<<vblk, 256, 0, stream>>>((v4f*)x1, n4);
    // 3) x1 = scatter(x0)
    k_scatter_edges<<<eblk, 256, 0, stream>>>(out, adj, w, x1, E);
    // 4) out = x0 + x1
    k_add4<<<vblk, 256, 0, stream>>>((v4f*)out, (const v4f*)x1, n4);
    // 5) out += scatter(x1)   (== x2)
    k_scatter_edges<<<eblk, 256, 0, stream>>>(x1, adj, w, out, E);
}